// OrdinalRegressionLoss_76312978915372
// MI455X (gfx1250) — compile-verified
//
#include <hip/hip_runtime.h>

typedef float v2f   __attribute__((ext_vector_type(2)));
typedef float v8f   __attribute__((ext_vector_type(8)));
typedef float f32x4 __attribute__((ext_vector_type(4)));

// Exact f32 sum of all 32 lanes, broadcast to every lane, using two
// V_WMMA_F32_16X16X4_F32 ops (B = all-ones).
//
// WMMA #1: A[m][k] = {v_m, 0, v_{m+16}, 0}  ->  D[m][n] = v_m + v_{m+16}
//          (lane-uniform across n). D VGPR j, lanes 0-15 = row j,
//          lanes 16-31 = row j+8 (ISA 7.12.2 C/D layout).
// Fold:    a0' = d0+d1+d2+d3, a1' = d4+d5+d6+d7  (lane-uniform per half)
// WMMA #2: rowsum = halves' partials summed -> every element of D' = total.
__device__ __forceinline__ float wave_sum_wmma(float v) {
    v2f a; a[0] = v;    a[1] = 0.0f;
    v2f b; b[0] = 1.0f; b[1] = 1.0f;
    v8f c = {};
    v8f d = __builtin_amdgcn_wmma_f32_16x16x4_f32(
        /*neg_a=*/false, a, /*neg_b=*/false, b,
        /*c_mod=*/(short)0, c, /*reuse_a=*/false, /*reuse_b=*/false);
    v2f a2;
    a2[0] = (d[0] + d[1]) + (d[2] + d[3]);
    a2[1] = (d[4] + d[5]) + (d[6] + d[7]);
    v8f d2 = __builtin_amdgcn_wmma_f32_16x16x4_f32(
        false, a2, false, b, (short)0, c, false, false);
    return d2[0];   // lane-uniform total
}

__device__ __forceinline__ float block_sum(float acc, int tid) {
    float ws = wave_sum_wmma(acc);
    __shared__ float smem[8];
    if ((tid & 31) == 0) smem[tid >> 5] = ws;
    __syncthreads();
    float s = 0.0f;
    if (tid == 0) {
        #pragma unroll
        for (int w = 0; w < 8; ++w) s += smem[w];
    }
    return s;   // valid on tid==0 only
}

// Stage 1: streaming partial reduction. One float4 logit load + one int
// label load per sample, grid-stride, non-temporal (single-pass data).
__global__ void __launch_bounds__(256)
ord_loss_partial(const float* __restrict__ logits,
                 const int*   __restrict__ labels,
                 float*       __restrict__ partial,
                 int B) {
    const int tid    = threadIdx.x;
    const int stride = gridDim.x * blockDim.x;
    float acc = 0.0f;
    for (int i = blockIdx.x * blockDim.x + tid; i < B; i += stride) {
        f32x4 x  = __builtin_nontemporal_load((const f32x4*)logits + i);
        int   lb = __builtin_nontemporal_load(labels + i);
        #pragma unroll
        for (int k = 0; k < 4; ++k) {
            float xv = x[k];
            float t  = (k < lb) ? xv : 0.0f;   // x * target
            // max(x,0) - x*t + log1p(exp(-|x|))
            acc += (fmaxf(xv, 0.0f) - t) + __logf(1.0f + __expf(-fabsf(xv)));
        }
    }
    float s = block_sum(acc, tid);     // EXEC all-1s here: no early exits
    if (tid == 0) partial[blockIdx.x] = s;
}

// Stage 2: single block folds the per-block partials and scales by 1/(B*K).
__global__ void __launch_bounds__(256)
ord_loss_final(const float* __restrict__ partial, int np,
               float* __restrict__ out, float inv_n) {
    const int tid = threadIdx.x;
    float acc = 0.0f;
    for (int i = tid; i < np; i += blockDim.x) acc += partial[i];
    float s = block_sum(acc, tid);
    if (tid == 0) out[0] = s * inv_n;
}

extern "C" void kernel_launch(void* const* d_in, const int* in_sizes, int n_in,
                              void* d_out, int out_size, void* d_ws, size_t ws_size,
                              hipStream_t stream) {
    (void)n_in; (void)out_size;
    const float* logits = (const float*)d_in[0];
    const int*   labels = (const int*)d_in[1];
    const int B = in_sizes[1];                 // number of samples
    const int K = (B > 0) ? (in_sizes[0] / B) : 4;   // = 4 logits/sample

    float* partial = (float*)d_ws;
    int NB = 4096;                             // saturates the device; 16 KB of ws
    size_t need = (size_t)NB * sizeof(float);
    if (need > ws_size) NB = (int)(ws_size / sizeof(float));
    if (NB < 1) NB = 1;

    ord_loss_partial<<<NB, 256, 0, stream>>>(logits, labels, partial, B);

    const float inv_n = (float)(1.0 / ((double)B * (double)K));  // 2^-25: exact
    ord_loss_final<<<1, 256, 0, stream>>>(partial, NB, (float*)d_out, inv_n);
}